// KnnBlock_79027398246590
// MI455X (gfx1250) — compile-verified
//
#include <hip/hip_runtime.h>

typedef __attribute__((ext_vector_type(2))) float v2f;
typedef __attribute__((ext_vector_type(8))) float v8f;

#define C_IN    64
#define NPIX    1024      // 32*32 patches per batch
#define KTOT    4096      // 64*8*8
#define NBATCH  4
#define KT      64        // K chunk staged in LDS
#define NT      16        // patches per block

// workspace layout (floats)
#define WS_S    0         // s[b*1024+p]
#define WS_T    4096      // t[b*1024+p]
#define WS_WSUM 8192      // 64
#define WS_BSUM 8256      // 1

// ---------------------------------------------------------------------------
// Fold the 1x1 conv + channel-sum:  sum_o(W2 h + b2) = (sum_o W2[o,:])·h + sum(b2)
__global__ void prep_kernel(const float* __restrict__ w2,
                            const float* __restrict__ b2,
                            float* __restrict__ ws) {
  int c = threadIdx.x;           // 64 threads
  float acc = 0.f;
  for (int o = 0; o < 64; ++o) acc += w2[o * 64 + c];
  ws[WS_WSUM + c] = acc;
  if (c == 0) {
    float s = 0.f;
    for (int o = 0; o < 64; ++o) s += b2[o];
    ws[WS_BSUM] = s;
  }
}

// ---------------------------------------------------------------------------
// Patch conv as GEMM (M=64 ch, K=4096, N=16 patches per block) with f32 WMMA,
// fused BN+ReLU+collapsed-1x1-channel-sum -> s, plus temperature head -> t.
__launch_bounds__(128)
__global__ void patch_gemm_kernel(const float* __restrict__ x,
                                  const float* __restrict__ w1,
                                  const float* __restrict__ b1,
                                  const float* __restrict__ g1,
                                  const float* __restrict__ bb1,
                                  const float* __restrict__ m1,
                                  const float* __restrict__ v1,
                                  const float* __restrict__ tw,
                                  const float* __restrict__ tb,
                                  const float* __restrict__ gt,
                                  const float* __restrict__ bt,
                                  const float* __restrict__ mt,
                                  const float* __restrict__ vt,
                                  float* __restrict__ ws) {
  __shared__ float Alds[64][KT + 1];      // weights: [co][k_local]
  __shared__ float Blds[KT][NT + 1];      // patch panel: [k_local][patch]
  __shared__ float sred[4][16];
  __shared__ float tred[128];

  const int tid  = threadIdx.x;
  const int lane = tid & 31;              // wave32
  const int wave = tid >> 5;              // 4 waves, one 16-channel M tile each
  const int b    = blockIdx.y;
  const int n0   = blockIdx.x * NT;

  // B-panel loading / temperature accumulation assignment
  const int pfix = tid & 15;              // patch slot owned by this thread
  const int klo  = tid >> 4;              // 0..7
  const int gp   = n0 + pfix;
  const int pi   = gp >> 5;
  const int pj   = gp & 31;
  const float* xb = x + (size_t)b * C_IN * 65536;

  // WMMA fragment coordinates (ISA 16x4 A / 4x16 B / 16x16 C layouts)
  const int mrow  = (wave << 4) + (lane & 15);
  const int khalf = (lane >> 4) << 1;     // lanes 0-15 -> K{0,1}, 16-31 -> K{2,3}
  const int col   = lane & 15;

  v8f acc = {0.f, 0.f, 0.f, 0.f, 0.f, 0.f, 0.f, 0.f};
  float tacc = 0.f;

  for (int kc = 0; kc < KTOT; kc += KT) {
    __syncthreads();
    // stage A chunk (64 x 64), coalesced along K
    #pragma unroll
    for (int e = 0; e < (64 * KT) / 128; ++e) {
      int idx = tid + e * 128;
      int m   = idx >> 6;
      int kk  = idx & 63;
      Alds[m][kk] = w1[m * KTOT + kc + kk];
    }
    // stage B chunk (64 x 16): k = ci*64 + ky*8 + kx maps straight into x
    #pragma unroll
    for (int s = 0; s < 8; ++s) {
      int kl = klo + (s << 3);
      int k  = kc + kl;
      int ci = k >> 6;
      int r  = k & 63;
      int ky = r >> 3;
      int kx = r & 7;
      Blds[kl][pfix] = xb[ci * 65536 + (pi * 8 + ky) * 256 + (pj * 8 + kx)];
    }
    __syncthreads();
    // temperature head reuses the staged patch panel (x read once from HBM)
    #pragma unroll
    for (int s = 0; s < 8; ++s) {
      int kl = klo + (s << 3);
      tacc += tw[kc + kl] * Blds[kl][pfix];
    }
    // exact fp32 matrix math: V_WMMA_F32_16X16X4_F32, K stepped by 4
    #pragma unroll
    for (int kk = 0; kk < KT; kk += 4) {
      int kb = kk + khalf;
      v2f af, bf;
      af.x = Alds[mrow][kb];
      af.y = Alds[mrow][kb + 1];
      bf.x = Blds[kb][col];
      bf.y = Blds[kb + 1][col];
      acc = __builtin_amdgcn_wmma_f32_16x16x4_f32(false, af, false, bf,
                                                  (short)0, acc, false, false);
    }
  }

  // epilogue: bias + BN + ReLU + collapsed channel-sum dot
  const float* wsum = ws + WS_WSUM;
  const int mhi = (lane >> 4) << 3;       // C/D VGPR r holds M=r (lo half) / r+8 (hi half)
  float partial = 0.f;
  #pragma unroll
  for (int r = 0; r < 8; ++r) {
    int co = (wave << 4) + mhi + r;
    float y = acc[r] + b1[co];
    float h = g1[co] * (y - m1[co]) * rsqrtf(v1[co] + 1e-5f) + bb1[co];
    h = fmaxf(h, 0.f);
    partial += wsum[co] * h;
  }
  partial += __shfl_xor(partial, 16, 32); // combine the two half-wave row groups
  if (lane < 16) sred[wave][lane] = partial;
  tred[tid] = tacc;
  __syncthreads();

  if (tid < 16) {
    float sv = sred[0][tid] + sred[1][tid] + sred[2][tid] + sred[3][tid]
             + ws[WS_BSUM];
    ws[WS_S + b * NPIX + n0 + tid] = sv;
    float tv = 0.f;
    #pragma unroll
    for (int wv = 0; wv < 8; ++wv) tv += tred[tid + (wv << 4)];
    tv += tb[0];
    tv = gt[0] * (tv - mt[0]) * rsqrtf(vt[0] + 1e-5f) + bt[0];
    ws[WS_T + b * NPIX + n0 + tid] = fmaxf(tv, 0.f);
  }
}

// ---------------------------------------------------------------------------
// One block per (batch,row-pixel): sim row lives in registers across the 4
// NN iterations; each output element hits HBM exactly once.
__launch_bounds__(256)
__global__ void softmax_kernel(const float* __restrict__ ws,
                               float* __restrict__ out) {
  const int row  = blockIdx.x;            // b*1024 + p
  const int b    = row >> 10;
  const int p    = row & 1023;
  const int tid  = threadIdx.x;
  const int lane = tid & 31;
  const int wid  = tid >> 5;
  __shared__ float red[8];

  const float* s  = ws + WS_S + b * NPIX;
  const float sp  = s[p];
  const float invT = 1.0f / (0.025f + ws[WS_T + row]);

  float sim[4];
  #pragma unroll
  for (int j = 0; j < 4; ++j) {
    int q = tid + (j << 8);
    float d = sp - s[q];
    sim[j] = -d * d + ((q == p) ? -1000.0f : 0.0f);
  }

  const size_t base = (size_t)row << 10;
  #pragma unroll
  for (int it = 0; it < 4; ++it) {
    float l[4];
    float lm = -3.402823466e38f;
    #pragma unroll
    for (int j = 0; j < 4; ++j) { l[j] = sim[j] * invT; lm = fmaxf(lm, l[j]); }
    #pragma unroll
    for (int o = 16; o > 0; o >>= 1) lm = fmaxf(lm, __shfl_xor(lm, o, 32));
    if (lane == 0) red[wid] = lm;
    __syncthreads();
    float m = red[0];
    #pragma unroll
    for (int k = 1; k < 8; ++k) m = fmaxf(m, red[k]);
    __syncthreads();

    float e[4], es = 0.f;
    #pragma unroll
    for (int j = 0; j < 4; ++j) { e[j] = __expf(l[j] - m); es += e[j]; }
    #pragma unroll
    for (int o = 16; o > 0; o >>= 1) es += __shfl_xor(es, o, 32);
    if (lane == 0) red[wid] = es;
    __syncthreads();
    float S = 0.f;
    #pragma unroll
    for (int k = 0; k < 8; ++k) S += red[k];
    float rS = 1.0f / S;

    float* op = out + (size_t)it * ((size_t)NBATCH * NPIX * NPIX) + base;
    #pragma unroll
    for (int j = 0; j < 4; ++j) {
      float pv = e[j] * rS;
      op[tid + (j << 8)] = pv;
      if (it < 3) sim[j] += __logf(1.0f - pv + 1e-45f);
    }
    __syncthreads();
  }
}

// ---------------------------------------------------------------------------
extern "C" void kernel_launch(void* const* d_in, const int* in_sizes, int n_in,
                              void* d_out, int out_size, void* d_ws, size_t ws_size,
                              hipStream_t stream) {
  (void)in_sizes; (void)n_in; (void)out_size; (void)ws_size;
  const float* x   = (const float*)d_in[0];
  const float* w1  = (const float*)d_in[1];
  const float* b1  = (const float*)d_in[2];
  const float* g1  = (const float*)d_in[3];
  const float* bb1 = (const float*)d_in[4];
  const float* m1  = (const float*)d_in[5];
  const float* v1  = (const float*)d_in[6];
  const float* w2  = (const float*)d_in[7];
  const float* b2  = (const float*)d_in[8];
  const float* tw  = (const float*)d_in[9];
  const float* tb  = (const float*)d_in[10];
  const float* gt  = (const float*)d_in[11];
  const float* bt  = (const float*)d_in[12];
  const float* mt  = (const float*)d_in[13];
  const float* vt  = (const float*)d_in[14];
  float* ws  = (float*)d_ws;
  float* out = (float*)d_out;

  prep_kernel<<<1, 64, 0, stream>>>(w2, b2, ws);

  dim3 gridA(NPIX / NT, NBATCH);          // 64 x 4 blocks, 128 threads (4 waves)
  patch_gemm_kernel<<<gridA, 128, 0, stream>>>(x, w1, b1, g1, bb1, m1, v1,
                                               tw, tb, gt, bt, mt, vt, ws);

  softmax_kernel<<<NBATCH * NPIX, 256, 0, stream>>>(ws, out);
}